// EdgeGraphConvolution_68813966016985
// MI455X (gfx1250) — compile-verified
//
#include <hip/hip_runtime.h>

// ---------------------------------------------------------------------------
// EdgeGraphConvolution, fused single-pass kernel for MI455X (gfx1250).
//
//   S[r,m]      = support[r,m] * (1/3) + (m == r/8 ? 1 : 0)   (written to d_out)
//   Y[16384,64] = S @ x                                        (WMMA f32 16x16x4)
//   Z           = relu(Y @ weight)                             (WMMA f32 16x16x4)
//
// Memory-bound: ~272 MB total HBM traffic (support in + support_mod out) ->
// support is streamed exactly once with non-temporal hints, transformed in
// registers, and forked to (a) the global support_mod output and (b) LDS
// staging for the WMMA A-operand. x/weight stay L2-resident (RT hints).
//
// B operands (x, weight) are staged pair-interleaved in LDS so each 4x16 B
// fragment (rows k,k+1 at one column) is a single aligned ds_load_b64 into an
// even VGPR pair -- no cross-register repacking in front of the WMMAs.
// ---------------------------------------------------------------------------

typedef __attribute__((ext_vector_type(2))) float v2f;
typedef __attribute__((ext_vector_type(4))) float v4f;
typedef __attribute__((ext_vector_type(8))) float v8f;

#define M_TOTAL 16384   // N*P rows of the flattened S matrix
#define K_TOTAL 2048    // N (contraction dim of the einsum)
#define DDIM    64      // D_IN == D_OUT
#define MT      64      // M rows per block (4 waves x 16)
#define KT      64      // K tile per iteration
#define PITCH   68      // sS row pitch (floats): (4r+c) mod 64 -> all 64 banks
                        // for the 32-lane b64 A-fragment read
#define PITCH2  160     // row-pair pitch (floats) for interleaved B staging:
                        // 160 mod 64 == 32 -> half-waves hit complementary banks

__global__ __launch_bounds__(128) void egc_fused_kernel(
    const float* __restrict__ x,        // [2048, 64]
    const float* __restrict__ support,  // [16384, 2048]
    const float* __restrict__ weight,   // [64, 64]
    float*       __restrict__ out_relu, // [16384, 64]   (d_out part 1)
    float*       __restrict__ out_smod) // [16384, 2048] (d_out part 2)
{
    __shared__ float sS [MT * PITCH];        // support_mod tile / Y stage (17.0 KB)
    __shared__ float sXp[(KT/2) * PITCH2];   // x slab, pair-interleaved   (20.0 KB)
    __shared__ float sWp[(DDIM/2) * PITCH2]; // weight, pair-interleaved   (20.0 KB)

    const int tid     = threadIdx.x;      // 0..127
    const int lane    = tid & 31;
    const int wave    = tid >> 5;         // 0..3
    const int halfSel = lane >> 4;        // 0: lanes 0-15, 1: lanes 16-31
    const int mrow    = lane & 15;        // row (A) / column (B,C,D) index
    const int waveRowBase = wave * 16;    // this wave's 16 rows within block
    const long R0     = (long)blockIdx.x * MT;

    // ---- weight -> LDS once, pair-interleaved: sWp[r/2][2*col + (r&1)] ----
    for (int i = tid; i < (DDIM * DDIM) / 4; i += 128) {
        int row = i >> 4;
        int c4  = (i & 15) << 2;
        v4f w = *(const v4f*)(weight + row * DDIM + c4);
        float* dst = &sWp[(row >> 1) * PITCH2 + (row & 1)];
        #pragma unroll
        for (int k = 0; k < 4; ++k) dst[2 * (c4 + k)] = w[k];
    }

    v8f acc[4];
    #pragma unroll
    for (int c = 0; c < 4; ++c)
        #pragma unroll
        for (int j = 0; j < 8; ++j) acc[c][j] = 0.0f;

    for (int kb = 0; kb < K_TOTAL; kb += KT) {
        // ---- cooperative: load support tile (NT), transform, fork ----------
        #pragma unroll
        for (int i = 0; i < (MT * KT) / (4 * 128); ++i) {   // 8 float4 / thread
            int idx = tid + i * 128;                        // 0..1023
            int row = idx >> 4;                             // 0..63
            int c4  = (idx & 15) << 2;                      // 0,4,...,60
            long gRow = R0 + row;
            long gOff = gRow * K_TOTAL + kb + c4;
            v4f v = __builtin_nontemporal_load((const v4f*)(support + gOff));
            v4f s = v * (1.0f / 3.0f);
            int rel = (int)(gRow >> 3) - (kb + c4);         // diag col = r / P
            if ((unsigned)rel < 4u) s[rel] += 1.0f;
            *(v4f*)(&sS[row * PITCH + c4]) = s;             // ds_store_b128
            __builtin_nontemporal_store(s, (v4f*)(out_smod + gOff));
        }
        // ---- cooperative: x K-slab -> LDS, pair-interleaved ----------------
        #pragma unroll
        for (int i = 0; i < (KT * DDIM) / (4 * 128); ++i) {
            int idx = tid + i * 128;
            int row = idx >> 4;
            int c4  = (idx & 15) << 2;
            v4f v = *(const v4f*)(x + (kb + row) * DDIM + c4);
            float* dst = &sXp[(row >> 1) * PITCH2 + (row & 1)];
            #pragma unroll
            for (int k = 0; k < 4; ++k) dst[2 * (c4 + k)] = v[k];
        }
        __syncthreads();

        // speculative prefetch of the next support tile (global_prefetch_b8)
        if (kb + KT < K_TOTAL) {
            const float* nxt =
                support + (R0 + (tid >> 1)) * K_TOTAL + (kb + KT) + (tid & 1) * 32;
            __builtin_prefetch(nxt, 0, 1);
        }

        // ---- WMMA: 16 K-steps x 4 column tiles per wave --------------------
        #pragma unroll 4
        for (int kk = 0; kk < KT; kk += 4) {
            // A 16x4 frag: lane row = waveRowBase+mrow, K = kk+2*halfSel+{0,1}
            v2f a = *(const v2f*)(
                &sS[(waveRowBase + mrow) * PITCH + kk + 2 * halfSel]);
            const int rp = (kk >> 1) + halfSel;             // row-pair index
            #pragma unroll
            for (int c = 0; c < 4; ++c) {
                // B 4x16 frag: one aligned b64 read, column = c*16 + mrow
                v2f b = *(const v2f*)(&sXp[rp * PITCH2 + 2 * (c * 16 + mrow)]);
                acc[c] = __builtin_amdgcn_wmma_f32_16x16x4_f32(
                    false, a, false, b, (short)0, acc[c], false, false);
            }
        }
        __syncthreads();
    }

    // ---- epilogue: Z = relu(Y @ W) ----
    // Stage Y (C/D layout -> row-major, reusing sS; each wave its own 16 rows)
    #pragma unroll
    for (int c = 0; c < 4; ++c)
        #pragma unroll
        for (int v = 0; v < 8; ++v) {
            int row = waveRowBase + v + 8 * halfSel;        // C/D: M = vgpr(+8)
            sS[row * PITCH + c * 16 + mrow] = acc[c][v];    // N = lane%16
        }
    __syncthreads();

    v8f z[4];
    #pragma unroll
    for (int c = 0; c < 4; ++c)
        #pragma unroll
        for (int j = 0; j < 8; ++j) z[c][j] = 0.0f;

    #pragma unroll 4
    for (int kk = 0; kk < DDIM; kk += 4) {
        v2f a = *(const v2f*)(
            &sS[(waveRowBase + mrow) * PITCH + kk + 2 * halfSel]);
        const int rp = (kk >> 1) + halfSel;
        #pragma unroll
        for (int c = 0; c < 4; ++c) {
            v2f b = *(const v2f*)(&sWp[rp * PITCH2 + 2 * (c * 16 + mrow)]);
            z[c] = __builtin_amdgcn_wmma_f32_16x16x4_f32(
                false, a, false, b, (short)0, z[c], false, false);
        }
    }

    // ReLU + store (each (c,v): half-wave writes one contiguous 64B row chunk)
    #pragma unroll
    for (int c = 0; c < 4; ++c)
        #pragma unroll
        for (int v = 0; v < 8; ++v) {
            long row = R0 + waveRowBase + v + 8 * halfSel;
            out_relu[row * DDIM + c * 16 + mrow] = fmaxf(z[c][v], 0.0f);
        }
}

extern "C" void kernel_launch(void* const* d_in, const int* in_sizes, int n_in,
                              void* d_out, int out_size, void* d_ws, size_t ws_size,
                              hipStream_t stream) {
    (void)in_sizes; (void)n_in; (void)d_ws; (void)ws_size; (void)out_size;
    const float* x       = (const float*)d_in[0];  // [2048, 64]
    const float* support = (const float*)d_in[1];  // [2048, 8, 2048]
    const float* weight  = (const float*)d_in[2];  // [64, 64]

    float* out_relu = (float*)d_out;                         // [2048,8,64]
    float* out_smod = (float*)d_out + (long)M_TOTAL * DDIM;  // [2048,8,2048]

    dim3 grid(M_TOTAL / MT);   // 256 blocks
    dim3 block(128);           // 4 waves (wave32)
    egc_fused_kernel<<<grid, block, 0, stream>>>(x, support, weight,
                                                 out_relu, out_smod);
}